// ReadHeadsModule_88261577933109
// MI455X (gfx1250) — compile-verified
//
#include <hip/hip_runtime.h>
#include <hip/hip_bf16.h>
#include <math.h>

// ---------------------------------------------------------------------------
// ReadHeadsModule for MI455X (gfx1250, wave32).
//
// Pipeline:
//  K1 sim_kernel    : masked cosine sim scan over key_memory (BW-bound, 512MB)
//  K2 topk_kernel   : per-batch top-16 + softmax + masked value gather -> X
//  K3 qkv_kernel    : X[1024,256] @ in_w^T -> Y[1024,768]  (V_WMMA_F32_16X16X4_F32)
//  K4 attn_kernel   : per-batch 16x16 softmax attention; S_b = sum_t O_t
//  K5 outproj_kernel: S[64,256] @ out_w^T + 16*b, masked     (fp32 WMMA)
// ---------------------------------------------------------------------------

typedef float v2f __attribute__((ext_vector_type(2)));
typedef float v8f __attribute__((ext_vector_type(8)));

#define N_MEM 8192
#define DK    256
#define DV    256
#define BATCH 64
#define TOPK  16

// ---------------- Kernel 1: cosine similarity scan (bandwidth-bound) -------
__global__ __launch_bounds__(256)
void sim_kernel(const float* __restrict__ q, const float* __restrict__ key,
                const int* __restrict__ iter, float* __restrict__ sims,
                int rows_per_block) {
    const int b     = blockIdx.x;
    const int chunk = blockIdx.y;
    const int lane  = threadIdx.x & 31;
    const int wave  = threadIdx.x >> 5;
    const int waves = blockDim.x >> 5;

    // each lane owns 8 contiguous query elements (2 x float4 = b128 loads)
    const float* qb = q + b * DK;
    const float4 q0 = *(const float4*)(qb + lane * 8);
    const float4 q1 = *(const float4*)(qb + lane * 8 + 4);

    float qn = q0.x*q0.x + q0.y*q0.y + q0.z*q0.z + q0.w*q0.w
             + q1.x*q1.x + q1.y*q1.y + q1.z*q1.z + q1.w*q1.w;
    #pragma unroll
    for (int off = 16; off; off >>= 1) qn += __shfl_xor(qn, off, 32);
    qn = sqrtf(qn);

    const int it   = iter[b];
    const int row0 = chunk * rows_per_block;
    const float* kb = key + (size_t)b * N_MEM * DK;

    for (int r = row0 + wave; r < row0 + rows_per_block; r += waves) {
        const float* krow = kb + (size_t)r * DK + lane * 8;
        // speculative prefetch ~2 iterations ahead (global_prefetch_b8)
        __builtin_prefetch(krow + 2 * waves * DK, 0, 3);
        const float4 k0 = *(const float4*)(krow);
        const float4 k1 = *(const float4*)(krow + 4);

        float dot = k0.x*q0.x + k0.y*q0.y + k0.z*q0.z + k0.w*q0.w
                  + k1.x*q1.x + k1.y*q1.y + k1.z*q1.z + k1.w*q1.w;
        float nsq = k0.x*k0.x + k0.y*k0.y + k0.z*k0.z + k0.w*k0.w
                  + k1.x*k1.x + k1.y*k1.y + k1.z*k1.z + k1.w*k1.w;
        #pragma unroll
        for (int off = 16; off; off >>= 1) {
            dot += __shfl_xor(dot, off, 32);
            nsq += __shfl_xor(nsq, off, 32);
        }
        if (lane == 0) {
            // masked row => km==0 => num=0, denom=eps => sim exactly 0
            float denom = fmaxf(sqrtf(nsq) * qn, 1e-8f);
            float s = (r <= it) ? (dot / denom) : 0.0f;
            sims[(size_t)b * N_MEM + r] = s;
        }
    }
}

// ---------------- Kernel 2: top-16 + softmax + masked gather ---------------
__global__ __launch_bounds__(256)
void topk_kernel(const float* __restrict__ sims, const float* __restrict__ vmem,
                 const int* __restrict__ iter, float* __restrict__ X) {
    __shared__ float s_sim[N_MEM];
    __shared__ float cand_v[256];
    __shared__ int   cand_i[256];
    __shared__ float top_v[TOPK];
    __shared__ int   top_i[TOPK];
    __shared__ float w[TOPK];

    const int b = blockIdx.x, tid = threadIdx.x;
    const float* sb = sims + (size_t)b * N_MEM;
    for (int i = tid; i < N_MEM; i += 256) s_sim[i] = sb[i];
    __syncthreads();

    // 16 argmax passes; ties -> lowest index (JAX top_k stability)
    for (int kk = 0; kk < TOPK; kk++) {
        float bv = -3e30f; int bi = 0;
        const int base = tid * 32;                // contiguous chunk per thread
        #pragma unroll 4
        for (int j = 0; j < 32; j++) {
            float v = s_sim[base + j];
            if (v > bv) { bv = v; bi = base + j; }
        }
        cand_v[tid] = bv; cand_i[tid] = bi;
        __syncthreads();
        if (tid == 0) {
            float best = cand_v[0]; int besti = cand_i[0];
            for (int t = 1; t < 256; t++)
                if (cand_v[t] > best) { best = cand_v[t]; besti = cand_i[t]; }
            top_v[kk] = best; top_i[kk] = besti;
            s_sim[besti] = -1e30f;                // exclude
        }
        __syncthreads();
    }

    if (tid == 0) {
        float mx = top_v[0];
        for (int k = 1; k < TOPK; k++) mx = fmaxf(mx, top_v[k]);
        float sum = 0.0f;
        for (int k = 0; k < TOPK; k++) { float e = expf(top_v[k] - mx); w[k] = e; sum += e; }
        float inv = 1.0f / sum;
        for (int k = 0; k < TOPK; k++) w[k] *= inv;
    }
    __syncthreads();

    const int it = iter[b];
    const float* vb = vmem + (size_t)b * N_MEM * DV;
    for (int i = tid; i < TOPK * DV; i += 256) {
        const int kk = i >> 8, d = i & 255;
        const int idx = top_i[kk];
        const float val = (idx <= it) ? vb[(size_t)idx * DV + d] : 0.0f;
        X[((size_t)b * TOPK + kk) * DV + d] = w[kk] * val;
    }
}

// ---------------- Kernel 3: Y = X @ in_w^T + in_b  (fp32 WMMA) -------------
// X:[1024,256]  in_w:[768,256]  Y:[1024,768]. One wave per 16x16 tile.
__global__ __launch_bounds__(256)
void qkv_kernel(const float* __restrict__ X, const float* __restrict__ inW,
                const float* __restrict__ inB, float* __restrict__ Y) {
    const int lane = threadIdx.x & 31;
    const int wave = threadIdx.x >> 5;
    const int tile = blockIdx.x * 8 + wave;       // 64*48 = 3072 tiles
    const int rbase = (tile / 48) * 16;           // row tile (tokens)
    const int cbase = (tile % 48) * 16;           // col tile (qkv dims)

    const int m    = lane & 15;
    const int koff = (lane & 16) ? 2 : 0;         // ISA 16x4 A / 4x16 B layout
    const float* xr = X   + (size_t)(rbase + m) * DK;
    const float* wr = inW + (size_t)(cbase + m) * DK;

    v8f acc = {0.f, 0.f, 0.f, 0.f, 0.f, 0.f, 0.f, 0.f};
    for (int k0 = 0; k0 < DK; k0 += 4) {
        v2f a, bb;
        a[0]  = xr[k0 + koff];  a[1]  = xr[k0 + koff + 1];
        bb[0] = wr[k0 + koff];  bb[1] = wr[k0 + koff + 1];
        acc = __builtin_amdgcn_wmma_f32_16x16x4_f32(
            false, a, false, bb, (short)0, acc, false, false);
    }

    const float bias  = inB[cbase + m];
    const int   half8 = (lane & 16) ? 8 : 0;
    #pragma unroll
    for (int r = 0; r < 8; r++) {
        const int row = rbase + r + half8;
        Y[(size_t)row * 768 + cbase + m] = acc[r] + bias;
    }
}

// ---------------- Kernel 4: per-batch attention, row-summed ----------------
// S_b[d] = sum_t (attn @ v)[t,d] = sum_s colsum(attn)[s] * v[s,d]
__global__ __launch_bounds__(256)
void attn_kernel(const float* __restrict__ Y, float* __restrict__ S) {
    __shared__ float qs[TOPK * DV];
    __shared__ float ks[TOPK * DV];
    __shared__ float attn[TOPK * TOPK];
    __shared__ float csum[TOPK];

    const int b = blockIdx.x, tid = threadIdx.x;
    const float* Yb = Y + (size_t)b * TOPK * 768;

    for (int i = tid; i < TOPK * DV; i += 256) {
        const int t = i >> 8, d = i & 255;
        qs[i] = Yb[t * 768 + d];
        ks[i] = Yb[t * 768 + 256 + d];
    }
    __syncthreads();

    {   // thread -> (t, s) score
        const int t = tid >> 4, s = tid & 15;
        float a = 0.0f;
        for (int d = 0; d < DV; d++) a += qs[t * DV + d] * ks[s * DV + d];
        attn[t * 16 + s] = a * 0.0625f;           // 1/sqrt(256)
    }
    __syncthreads();

    if (tid < TOPK) {                              // row softmax
        float mx = -3e30f;
        for (int s = 0; s < 16; s++) mx = fmaxf(mx, attn[tid * 16 + s]);
        float sum = 0.0f;
        for (int s = 0; s < 16; s++) { float e = expf(attn[tid * 16 + s] - mx); attn[tid * 16 + s] = e; sum += e; }
        float inv = 1.0f / sum;
        for (int s = 0; s < 16; s++) attn[tid * 16 + s] *= inv;
    }
    __syncthreads();
    if (tid < TOPK) {                              // column sums
        float c = 0.0f;
        for (int t = 0; t < 16; t++) c += attn[t * 16 + tid];
        csum[tid] = c;
    }
    __syncthreads();

    {   // S[d] = sum_s csum[s] * v[s,d]
        const int d = tid;
        float a = 0.0f;
        #pragma unroll
        for (int s = 0; s < TOPK; s++) a += csum[s] * Yb[s * 768 + 512 + d];
        S[(size_t)b * DV + d] = a;
    }
}

// ---------------- Kernel 5: out = mask * (S @ out_w^T + 16*out_b) ----------
// S:[64,256]  out_w:[256,256]  out:[64,256]. fp32 WMMA, 4x16 = 64 tiles.
__global__ __launch_bounds__(256)
void outproj_kernel(const float* __restrict__ S, const float* __restrict__ oW,
                    const float* __restrict__ oB, const int* __restrict__ iter,
                    float* __restrict__ out) {
    const int lane = threadIdx.x & 31;
    const int wave = threadIdx.x >> 5;
    const int tile = blockIdx.x * 8 + wave;       // 64 tiles total
    const int rbase = (tile >> 4) * 16;           // batch rows
    const int cbase = (tile & 15) * 16;           // output dims

    const int m    = lane & 15;
    const int koff = (lane & 16) ? 2 : 0;
    const float* sr = S  + (size_t)(rbase + m) * DV;
    const float* wr = oW + (size_t)(cbase + m) * DV;

    v8f acc = {0.f, 0.f, 0.f, 0.f, 0.f, 0.f, 0.f, 0.f};
    for (int k0 = 0; k0 < DV; k0 += 4) {
        v2f a, bb;
        a[0]  = sr[k0 + koff];  a[1]  = sr[k0 + koff + 1];
        bb[0] = wr[k0 + koff];  bb[1] = wr[k0 + koff + 1];
        acc = __builtin_amdgcn_wmma_f32_16x16x4_f32(
            false, a, false, bb, (short)0, acc, false, false);
    }

    const float bias  = 16.0f * oB[cbase + m];    // sum over T=16 rows
    const int   half8 = (lane & 16) ? 8 : 0;
    #pragma unroll
    for (int r = 0; r < 8; r++) {
        const int row = rbase + r + half8;        // batch index
        const float v = acc[r] + bias;
        out[(size_t)row * DV + cbase + m] = (iter[row] > 0) ? v : 0.0f;
    }
}

// ---------------------------------------------------------------------------
extern "C" void kernel_launch(void* const* d_in, const int* in_sizes, int n_in,
                              void* d_out, int out_size, void* d_ws, size_t ws_size,
                              hipStream_t stream) {
    const float* query = (const float*)d_in[0];   // [64,256]
    const float* keym  = (const float*)d_in[1];   // [64,8192,256]
    const float* valm  = (const float*)d_in[2];   // [64,8192,256]
    const int*   iter  = (const int*)  d_in[3];   // [64,1]
    const float* inW   = (const float*)d_in[4];   // [768,256]
    const float* inB   = (const float*)d_in[5];   // [768]
    const float* oW    = (const float*)d_in[6];   // [256,256]
    const float* oB    = (const float*)d_in[7];   // [256]
    float* out = (float*)d_out;                   // [64,256]

    float* ws   = (float*)d_ws;
    float* sims = ws;                             // 64*8192      = 524288 f
    float* X    = ws + 524288;                    // 1024*256     = 262144 f
    float* Y    = ws + 786432;                    // 1024*768     = 786432 f
    float* S    = ws + 1572864;                   // 64*256       =  16384 f

    // K1: 64 batches x 32 chunks of 256 rows each
    sim_kernel<<<dim3(BATCH, 32), 256, 0, stream>>>(query, keym, iter, sims, 256);
    // K2: one block per batch
    topk_kernel<<<BATCH, 256, 0, stream>>>(sims, valm, iter, X);
    // K3: 3072 tiles / 8 waves per block
    qkv_kernel<<<384, 256, 0, stream>>>(X, inW, inB, Y);
    // K4: one block per batch
    attn_kernel<<<BATCH, 256, 0, stream>>>(Y, S);
    // K5: 64 tiles / 8 waves per block
    outproj_kernel<<<8, 256, 0, stream>>>(S, oW, oB, iter, out);
}